// SingleLayerDecoder_27968827032007
// MI455X (gfx1250) — compile-verified
//
#include <hip/hip_runtime.h>
#include <hip/hip_bf16.h>
#include <math.h>

// Problem constants (reference: B=8, L=256, D=512, H=512)
#define BB   8
#define LL   256
#define DD   512
#define HH   512
#define MM   (BB * LL)               // 2048 rows in the fused [B*L, D] GEMM
#define NOUT ((size_t)LL * BB * LL)  // 524288 elements per output tensor

#define CH   16                      // h-rows per async-staged chunk (16KB)
#define NC   (HH / CH)               // 32 chunks

typedef __attribute__((ext_vector_type(2))) float v2f;
typedef __attribute__((ext_vector_type(8))) float v8f;

// ---------------------------------------------------------------------------
// GEMM: C[M,N] = X[M,K] * W[K,N] with f32 WMMA 16x16x4.
// Each wave computes one 16x16 tile; 8 waves (256 thr) per block -> 8 tiles.
// If transposed_store: D tile for row-group (b,j) x col-group (h) is written
// into drT laid out as [B][H][L] (so the pairwise kernel reads coalesced).
// ---------------------------------------------------------------------------
__global__ __launch_bounds__(256) void wmma_gemm_f32(
    const float* __restrict__ X,   // [MM, DD]
    const float* __restrict__ W,   // [DD, HH]
    float* __restrict__ C,         // [MM, HH]  or  [BB, HH, LL] if transposed
    int transposed_store)
{
    const int tid   = threadIdx.x;
    const int lane  = tid & 31;
    const int wave  = tid >> 5;
    const int half  = lane >> 4;     // 0: K=0,1 (A) / M+=0 (D);  1: K=2,3 / M+=8
    const int l16   = lane & 15;

    const int tile  = blockIdx.x * 8 + wave;       // 4096 tiles total
    const int tn    = tile & ((HH / 16) - 1);      // 32 tiles along N
    const int tm    = tile >> 5;                   // 128 tiles along M
    const int m0    = tm * 16;
    const int n0    = tn * 16;

    const float* xrow = X + (size_t)(m0 + l16) * DD;
    const int    ncol = n0 + l16;

    v8f acc = {};  // C starts at zero

    #pragma unroll 4
    for (int k = 0; k < DD; k += 4) {
        const int kk = k + 2 * half;
        v2f a = *(const v2f*)(xrow + kk);                 // A[m][kk], A[m][kk+1]
        v2f b;
        b.x = W[(size_t)kk       * HH + ncol];            // B[kk][n]
        b.y = W[(size_t)(kk + 1) * HH + ncol];            // B[kk+1][n]
        acc = __builtin_amdgcn_wmma_f32_16x16x4_f32(
                  /*neg_a=*/false, a, /*neg_b=*/false, b,
                  /*c_mod=*/(short)0, acc,
                  /*reuse_a=*/false, /*reuse_b=*/false);
    }

    if (!transposed_store) {
        // D layout: VGPR v holds (M = m0 + v + 8*half, N = n0 + l16)
        #pragma unroll
        for (int v = 0; v < 8; ++v)
            C[(size_t)(m0 + v + 8 * half) * HH + ncol] = acc[v];
    } else {
        // Row index m encodes (b, j); tiles never cross b (256 % 16 == 0).
        const int b_idx = m0 >> 8;          // m0 / LL
        const int jbase = m0 & (LL - 1);
        float* dst = C + (size_t)b_idx * HH * LL + (size_t)ncol * LL;
        #pragma unroll
        for (int v = 0; v < 8; ++v)
            dst[jbase + v + 8 * half] = acc[v];   // drT[b][h=ncol][j]
    }
}

// ---------------------------------------------------------------------------
// Pairwise biaffine + tanh + U-reduction + sigmoid/bernoulli/entropy.
// One block per (b,i); thread t owns column j = t. drT is streamed through a
// double-buffered LDS staging pipeline via GLOBAL_LOAD_ASYNC_TO_LDS_B128.
// ---------------------------------------------------------------------------
__device__ __forceinline__ float hw_tanh(float x) {
    float y;
    // CDNA5 TRANS op; v_nop covers the TRANS result-use hazard.
    asm volatile("v_tanh_f32 %0, %1\n\tv_nop" : "=v"(y) : "v"(x));
    return y;
}

__device__ __forceinline__ unsigned pcg_hash(unsigned x) {
    x = x * 747796405u + 2891336453u;
    x = ((x >> ((x >> 28) + 4u)) ^ x) * 277803737u;
    return (x >> 22) ^ x;
}

// Issue 4 x 16B async copies: one contiguous 16KB chunk, 256 threads.
__device__ __forceinline__ void async_chunk_copy(unsigned lds_off,
                                                 const float* gsrc) {
    #pragma unroll
    for (int k = 0; k < 4; ++k) {
        asm volatile("global_load_async_to_lds_b128 %0, %1, off"
                     :: "v"(lds_off + 4096u * k), "v"(gsrc + 1024 * k)
                     : "memory");
    }
}

__global__ __launch_bounds__(256) void biaffine_pairwise(
    const float* __restrict__ dl,    // [MM, HH]       (dot_l)
    const float* __restrict__ drT,   // [BB, HH, LL]   (dot_r transposed)
    const float* __restrict__ U,     // [HH]
    const float* __restrict__ bias,  // [1]
    float* __restrict__ out)         // [3*NOUT]: samples(i32 bits)|scores|entropy
{
    __shared__ float s_dl[HH];
    __shared__ float s_U[HH];
    __shared__ float __align__(16) s_buf[2][CH * LL];   // 2 x 16KB double buffer

    const int blk = blockIdx.x;          // 2048 = B*L blocks
    const int b   = blk >> 8;            // blk / LL
    const int i   = blk & (LL - 1);
    const int t   = threadIdx.x;         // 256 threads, j = t

    s_dl[t]       = dl[(size_t)blk * HH + t];
    s_dl[t + LL]  = dl[(size_t)blk * HH + t + LL];
    s_U[t]        = U[t];
    s_U[t + LL]   = U[t + LL];

    const float* drb = drT + (size_t)b * HH * LL;

    // This thread's 16B-aligned destination slots inside each LDS buffer.
    const unsigned lds0 = (unsigned)(uintptr_t)(&s_buf[0][0]) + (unsigned)t * 16u;
    const unsigned lds1 = (unsigned)(uintptr_t)(&s_buf[1][0]) + (unsigned)t * 16u;

    // Prefetch chunk 0 into buffer 0.
    async_chunk_copy(lds0, drb + t * 4);

    const int j = t;
    float s = 0.0f;

    for (int c = 0; c < NC; ++c) {
        asm volatile("s_wait_asynccnt 0x0" ::: "memory");  // my chunk-c copies done
        __syncthreads();  // everyone's chunk c in LDS; everyone done with chunk c-1

        if (c + 1 < NC) {
            // Other buffer was last read during chunk c-1 -> safe to overwrite.
            async_chunk_copy(((c + 1) & 1) ? lds1 : lds0,
                             drb + (size_t)(c + 1) * (CH * LL) + t * 4);
        }

        const float* bufc = &s_buf[c & 1][0];
        const int h0 = c * CH;
        #pragma unroll
        for (int hh = 0; hh < CH; ++hh) {
            float x = s_dl[h0 + hh] + bufc[hh * LL + j];   // conflict-free LDS
            s = fmaf(s_U[h0 + hh], hw_tanh(x), s);
        }
    }

    s += bias[0];
    if (j == i) s -= 1.0e8f;             // self-mask (NEG_BIG * eye)

    // sigmoid + numerically-stable softplus; entropy = softplus(s) - p*s
    const float p   = 1.0f / (1.0f + __expf(-s));
    const float sp  = fmaxf(s, 0.0f) + log1pf(__expf(-fabsf(s)));
    const float ent = sp - p * s;

    // mask_scores layout is [L, B, L] (transpose(1,0,2) in the reference)
    const size_t idx = (size_t)i * (BB * LL) + (size_t)b * LL + (size_t)j;

    // bernoulli draw (hash-based; JAX threefry not reproducible here)
    unsigned hsh = pcg_hash((unsigned)idx ^ 0xDEADBE42u);
    const float u = (float)(hsh >> 8) * (1.0f / 16777216.0f);
    const int smp = (u < p) ? 1 : 0;

    ((int*)out)[idx]    = smp;   // samples  (int32)
    out[NOUT + idx]     = s;     // mask_scores
    out[2 * NOUT + idx] = ent;   // entropy
}

// ---------------------------------------------------------------------------
extern "C" void kernel_launch(void* const* d_in, const int* in_sizes, int n_in,
                              void* d_out, int out_size, void* d_ws, size_t ws_size,
                              hipStream_t stream) {
    const float* enc  = (const float*)d_in[0];   // [B, L, D]
    const float* W_l  = (const float*)d_in[1];   // [D, H]
    const float* W_r  = (const float*)d_in[2];   // [D, H]
    const float* U    = (const float*)d_in[3];   // [H]
    const float* bias = (const float*)d_in[4];   // [1]

    float* dl  = (float*)d_ws;                       // 4 MB: [MM, HH]
    float* drT = (float*)d_ws + (size_t)MM * HH;     // 4 MB: [BB, HH, LL]

    const int gemm_blocks = (MM / 16) * (HH / 16) / 8;   // 4096 tiles / 8 per block
    wmma_gemm_f32<<<gemm_blocks, 256, 0, stream>>>(enc, W_l, dl, 0);
    wmma_gemm_f32<<<gemm_blocks, 256, 0, stream>>>(enc, W_r, drT, 1);

    biaffine_pairwise<<<BB * LL, 256, 0, stream>>>(dl, drT, U, bias, (float*)d_out);
}